// DepthPriorLoss_69793218560287
// MI455X (gfx1250) — compile-verified
//
#include <hip/hip_runtime.h>
#include <hip/hip_bf16.h>
#include <cstdint>

// ---------------------------------------------------------------------------
// DepthPriorLoss for MI455X (gfx1250, wave32)
//  - elementwise passes: memory-bound (~32MB -> ~1.4us @ 23.3 TB/s), float4 loads
//  - median/MAD: 4096-bin histograms, int atomics (deterministic)
//  - RANSAC residual matrix recast as (1000x4)x(4x50000) f32 GEMM using
//    V_WMMA_F32_16X16X4_F32, 2x unrolled; B staged to LDS via
//    GLOBAL_LOAD_ASYNC_TO_LDS_B128 (ASYNCcnt) when available
// ---------------------------------------------------------------------------

#define NBINS 4096
#define HIST_RANGE 10.0f          // y = 1/(d+1e-6), d>0.1 => y < 10
#define N_ITER 1000
#define N_ITER_PAD 1008           // 63 tiles of 16
#define N_SUB 50000               // 3125 tiles of 16 (exact)
#define COL_CHUNKS 64
#define TPC ((N_SUB / 16 + COL_CHUNKS - 1) / COL_CHUNKS)   // 49 col-tiles/chunk
#define THRESH_C 0.01f

typedef float v2f __attribute__((ext_vector_type(2)));
typedef float v8f __attribute__((ext_vector_type(8)));
typedef int   v4i __attribute__((__vector_size__(4 * sizeof(int))));

#if defined(__has_builtin)
#if __has_builtin(__builtin_amdgcn_global_load_async_to_lds_b128) && \
    __has_builtin(__builtin_amdgcn_s_wait_asynccnt)
#define HAVE_ASYNC_LDS 1
#endif
#endif

__device__ __forceinline__ uint32_t hash32(uint32_t x) {
    x ^= x >> 16; x *= 0x7feb352du;
    x ^= x >> 15; x *= 0x846ca68bu;
    x ^= x >> 16;
    return x;
}

__device__ __forceinline__ bool mask_of(float d) {
    return (d > 0.1f) && (d < 100.0f) && isfinite(d);
}

__device__ __forceinline__ int probe_masked(const float* __restrict__ depth,
                                            int HW, uint32_t h) {
    int idx = (int)(h % (uint32_t)HW);
    for (int k = 0; k < 1024; ++k) {
        if (mask_of(depth[idx])) return idx;
        ++idx; if (idx >= HW) idx = 0;
    }
    return idx;
}

// wave32 xor-butterfly sum within each 16-lane half (ISA SWAPX1/2/4/8)
__device__ __forceinline__ int half_wave_sum(int v) {
    v += __builtin_amdgcn_ds_swizzle(v, 0x1f | (1 << 10));
    v += __builtin_amdgcn_ds_swizzle(v, 0x1f | (2 << 10));
    v += __builtin_amdgcn_ds_swizzle(v, 0x1f | (4 << 10));
    v += __builtin_amdgcn_ds_swizzle(v, 0x1f | (8 << 10));
    return v;
}

// ---------------------------------------------------------------------------
// Pass 1: target_inv = 1/(d+1e-6), n_valid count, histogram of y (masked)
// float4 loads of depth (aligned); scalar stores (out region is +4B offset)
// ---------------------------------------------------------------------------
__global__ __launch_bounds__(256) void k_pass1(
    const float* __restrict__ depth, float* __restrict__ target_inv,
    unsigned* __restrict__ hist, unsigned* __restrict__ nValid, int HW)
{
    __shared__ unsigned lhist[NBINS];
    __shared__ unsigned lcount;
    for (int i = threadIdx.x; i < NBINS; i += blockDim.x) lhist[i] = 0u;
    if (threadIdx.x == 0) lcount = 0u;
    __syncthreads();

    const float4* __restrict__ d4 = (const float4*)depth;
    const int N4 = HW >> 2;
    const int stride = gridDim.x * blockDim.x;
    unsigned my = 0;
    for (int i = blockIdx.x * blockDim.x + threadIdx.x; i < N4; i += stride) {
        float4 dv = d4[i];
        float dc[4] = {dv.x, dv.y, dv.z, dv.w};
#pragma unroll
        for (int c = 0; c < 4; ++c) {
            float d = dc[c];
            float y = 1.0f / (d + 1e-6f);
            target_inv[4 * i + c] = y;
            if (mask_of(d)) {
                ++my;
                int b = (int)(y * (NBINS / HIST_RANGE));
                b = min(max(b, 0), NBINS - 1);
                atomicAdd(&lhist[b], 1u);
            }
        }
    }
    atomicAdd(&lcount, my);
    __syncthreads();
    for (int i = threadIdx.x; i < NBINS; i += blockDim.x)
        if (lhist[i]) atomicAdd(&hist[i], lhist[i]);
    if (threadIdx.x == 0) atomicAdd(nValid, lcount);
}

// ---------------------------------------------------------------------------
// Histogram median scan (single thread; 4096 iterations, negligible)
// ---------------------------------------------------------------------------
__global__ void k_quantile(const unsigned* __restrict__ hist,
                           const unsigned* __restrict__ nValid,
                           float* __restrict__ out, int isMad)
{
    if (threadIdx.x != 0 || blockIdx.x != 0) return;
    unsigned n = *nValid;
    unsigned half = (n + 1u) >> 1;
    unsigned c = 0; int bin = 0;
    for (bin = 0; bin < NBINS; ++bin) { c += hist[bin]; if (c >= half && half > 0) break; }
    if (bin >= NBINS) bin = NBINS - 1;
    float v = (bin + 0.5f) * (HIST_RANGE / NBINS);
    if (isMad) {
        float dyn = v * 0.5f;
        if (dyn < 1e-5f) dyn = THRESH_C;
        *out = dyn;
    } else {
        *out = v;
    }
}

// ---------------------------------------------------------------------------
// Pass 2: histogram of |y - med| over masked pixels (float4 loads)
// ---------------------------------------------------------------------------
__global__ __launch_bounds__(256) void k_pass2(
    const float* __restrict__ depth, const float* __restrict__ medp,
    unsigned* __restrict__ hist, int HW)
{
    __shared__ unsigned lhist[NBINS];
    for (int i = threadIdx.x; i < NBINS; i += blockDim.x) lhist[i] = 0u;
    __syncthreads();

    const float med = medp[0];
    const float4* __restrict__ d4 = (const float4*)depth;
    const int N4 = HW >> 2;
    const int stride = gridDim.x * blockDim.x;
    for (int i = blockIdx.x * blockDim.x + threadIdx.x; i < N4; i += stride) {
        float4 dv = d4[i];
        float dc[4] = {dv.x, dv.y, dv.z, dv.w};
#pragma unroll
        for (int c = 0; c < 4; ++c) {
            float d = dc[c];
            if (mask_of(d)) {
                float y = 1.0f / (d + 1e-6f);
                float ad = fabsf(y - med);
                int b = (int)(ad * (NBINS / HIST_RANGE));
                b = min(max(b, 0), NBINS - 1);
                atomicAdd(&lhist[b], 1u);
            }
        }
    }
    __syncthreads();
    for (int i = threadIdx.x; i < NBINS; i += blockDim.x)
        if (lhist[i]) atomicAdd(&hist[i], lhist[i]);
}

// ---------------------------------------------------------------------------
// RANSAC sampling: 1000 random pairs -> (scale, shift); pad to 1008
// ---------------------------------------------------------------------------
__global__ __launch_bounds__(256) void k_sample_pairs(
    const float* __restrict__ depth, const float* __restrict__ prior,
    float* __restrict__ scales, float* __restrict__ shifts, int HW)
{
    int i = blockIdx.x * blockDim.x + threadIdx.x;
    if (i >= N_ITER_PAD) return;
    if (i >= N_ITER) { scales[i] = 0.0f; shifts[i] = 0.0f; return; }
    uint32_t h1 = hash32(0x9E3779B9u ^ (uint32_t)(2 * i));
    uint32_t h2 = hash32(0x85EBCA6Bu ^ (uint32_t)(2 * i + 1));
    int i1 = probe_masked(depth, HW, h1);
    int i2 = probe_masked(depth, HW, h2);
    float x1 = prior[i1], x2 = prior[i2];
    float y1 = 1.0f / (depth[i1] + 1e-6f);
    float y2 = 1.0f / (depth[i2] + 1e-6f);
    float s = (y2 - y1) / (x2 - x1 + 1e-8f);
    float t = y1 - s * x1;
    scales[i] = s; shifts[i] = t;
}

// 50000 random subsample points -> x_sub, y_sub
__global__ __launch_bounds__(256) void k_sample_sub(
    const float* __restrict__ depth, const float* __restrict__ prior,
    float* __restrict__ xs, float* __restrict__ ys, int HW)
{
    int j = blockIdx.x * blockDim.x + threadIdx.x;
    if (j >= N_SUB) return;
    uint32_t h = hash32(0xC0FFEE11u ^ (uint32_t)j);
    int idx = probe_masked(depth, HW, h);
    xs[j] = prior[idx];
    ys[j] = 1.0f / (depth[idx] + 1e-6f);
}

// ---------------------------------------------------------------------------
// Inlier counting via V_WMMA_F32_16X16X4_F32, 2x unrolled.
//   res[16x16] = A(16x4) * B(4x16),  A row i = (s_i, t_i, -1, 0),
//   B col j = (x_j, 1, y_j, 0)^T   => res = s*x + t - y
// B slice staged global->LDS with async-to-LDS b128 copies (ASYNCcnt),
// then fed from LDS (ds_load) to the WMMA pipe.
// ---------------------------------------------------------------------------
__global__ __launch_bounds__(32) void k_ransac_count(
    const float* __restrict__ scales, const float* __restrict__ shifts,
    const float* __restrict__ xs, const float* __restrict__ ys,
    const float* __restrict__ dynp, int* __restrict__ counts)
{
    __shared__ __align__(16) float lx[TPC * 16];
    __shared__ __align__(16) float ly[TPC * 16];

    const int lane = threadIdx.x;        // 0..31
    const int tile = blockIdx.x;         // iteration tile (63)
    const int chunk = blockIdx.y;        // subsample chunk (COL_CHUNKS)
    const float dyn = dynp[0];

    // A fragment: rows tile*16 .. tile*16+15
    // 32-bit A 16x4 layout: lanes 0-15 hold K=0,1; lanes 16-31 hold K=2,3.
    const int row = tile * 16 + (lane & 15);
    v2f a;
    if (lane < 16) { a.x = scales[row]; a.y = shifts[row]; }
    else           { a.x = -1.0f;       a.y = 0.0f;        }

    const int nTiles = N_SUB / 16;                    // 3125
    const int t0 = chunk * TPC;
    const int nT = min(TPC, nTiles - t0);             // 49 (last chunk: 38)
    const int nCols = nT * 16;
    const float* __restrict__ gx = xs + t0 * 16;
    const float* __restrict__ gy = ys + t0 * 16;

#ifdef HAVE_ASYNC_LDS
    {
        const int nVec = nCols >> 2;                  // 16B packets
        for (int i = lane; i < nVec; i += 32) {
            __builtin_amdgcn_global_load_async_to_lds_b128(
                (v4i*)(gx + 4 * i), (v4i*)&lx[4 * i], 0, 0);
            __builtin_amdgcn_global_load_async_to_lds_b128(
                (v4i*)(gy + 4 * i), (v4i*)&ly[4 * i], 0, 0);
        }
        __builtin_amdgcn_s_wait_asynccnt(0);          // drain ASYNCcnt
    }
#else
    for (int i = lane; i < nCols; i += 32) { lx[i] = gx[i]; ly[i] = gy[i]; }
#endif
    // single wave per block: LDS ops from this wave are in-order, no barrier

    int acc[8];
#pragma unroll
    for (int r = 0; r < 8; ++r) acc[r] = 0;

    const int n = lane & 15;
    const float bY  = (lane < 16) ? 1.0f : 0.0f;
    const float* __restrict__ lsrc = (lane < 16) ? lx : ly;

    int ct = 0;
    for (; ct + 2 <= nT; ct += 2) {                   // 2 WMMAs in flight
        v2f b0, b1;
        b0.x = lsrc[ct * 16 + n];      b0.y = bY;
        b1.x = lsrc[ct * 16 + 16 + n]; b1.y = bY;
        v8f c0 = {}, c1 = {};
        c0 = __builtin_amdgcn_wmma_f32_16x16x4_f32(
                false, a, false, b0, (short)0, c0, false, false);
        c1 = __builtin_amdgcn_wmma_f32_16x16x4_f32(
                false, a, false, b1, (short)0, c1, false, false);
#pragma unroll
        for (int r = 0; r < 8; ++r) {
            acc[r] += (fabsf(c0[r]) < dyn) ? 1 : 0;
            acc[r] += (fabsf(c1[r]) < dyn) ? 1 : 0;
        }
    }
    for (; ct < nT; ++ct) {                           // tail tile
        v2f b;
        b.x = lsrc[ct * 16 + n]; b.y = bY;
        v8f c = {};
        c = __builtin_amdgcn_wmma_f32_16x16x4_f32(
                false, a, false, b, (short)0, c, false, false);
#pragma unroll
        for (int r = 0; r < 8; ++r)
            acc[r] += (fabsf(c[r]) < dyn) ? 1 : 0;
    }

    // Reduce over N within each 16-lane half (ds_swizzle XOR butterflies)
#pragma unroll
    for (int r = 0; r < 8; ++r) acc[r] = half_wave_sum(acc[r]);

    // C/D layout: VGPR r -> M=r (lanes 0-15), M=r+8 (lanes 16-31)
    if ((lane & 15) == 0) {
        const int base = tile * 16 + ((lane == 16) ? 8 : 0);
#pragma unroll
        for (int r = 0; r < 8; ++r) {
            const int rr = base + r;
            if (rr < N_ITER) atomicAdd(&counts[rr], acc[r]);
        }
    }
}

// ---------------------------------------------------------------------------
// argmax over counts (scales<=0 => -1), fallback rules, write s_best/t_best
// ---------------------------------------------------------------------------
__global__ __launch_bounds__(256) void k_select(
    const float* __restrict__ scales, const float* __restrict__ shifts,
    const int* __restrict__ counts, const unsigned* __restrict__ nValid,
    float* __restrict__ sb, float* __restrict__ tb)
{
    __shared__ int sc[256];
    __shared__ int si[256];
    int bc = -2, bi = 0x7fffffff;
    for (int i = threadIdx.x; i < N_ITER; i += 256) {
        int c = (scales[i] > 0.0f) ? counts[i] : -1;
        if (c > bc || (c == bc && i < bi)) { bc = c; bi = i; }
    }
    sc[threadIdx.x] = bc; si[threadIdx.x] = bi;
    __syncthreads();
    for (int w = 128; w > 0; w >>= 1) {
        if (threadIdx.x < w) {
            int oc = sc[threadIdx.x + w], oi = si[threadIdx.x + w];
            if (oc > sc[threadIdx.x] ||
                (oc == sc[threadIdx.x] && oi < si[threadIdx.x])) {
                sc[threadIdx.x] = oc; si[threadIdx.x] = oi;
            }
        }
        __syncthreads();
    }
    if (threadIdx.x == 0) {
        bool valid = sc[0] >= 0;
        float s = valid ? scales[si[0]] : 1.0f;
        float t = valid ? shifts[si[0]] : 0.0f;
        unsigned n = *nValid;
        if (n < 10u) { s = 1.0f; t = 0.0f; }
        sb[0] = s; tb[0] = t;
    }
}

// ---------------------------------------------------------------------------
// Pass 3: aligned = s*x+t; per-block partial L1 sums; prior_metric_depth
// (recomputes y from depth: saves re-reading target_inv = 8MB of traffic)
// ---------------------------------------------------------------------------
__global__ __launch_bounds__(256) void k_pass3(
    const float* __restrict__ depth, const float* __restrict__ prior,
    const float* __restrict__ sb, const float* __restrict__ tb,
    float* __restrict__ pmd, float* __restrict__ blockSums, int HW)
{
    __shared__ float red[256];
    const float s = sb[0], t = tb[0];
    const float4* __restrict__ d4 = (const float4*)depth;
    const float4* __restrict__ p4 = (const float4*)prior;
    const int N4 = HW >> 2;
    const int stride = gridDim.x * blockDim.x;
    float acc = 0.0f;
    for (int i = blockIdx.x * blockDim.x + threadIdx.x; i < N4; i += stride) {
        float4 dv = d4[i];
        float4 xv = p4[i];
        float dc[4] = {dv.x, dv.y, dv.z, dv.w};
        float xc[4] = {xv.x, xv.y, xv.z, xv.w};
#pragma unroll
        for (int c = 0; c < 4; ++c) {
            float d = dc[c];
            float aligned = s * xc[c] + t;
            float y = 1.0f / (d + 1e-6f);
            if (mask_of(d)) acc += fabsf(aligned - y);
            pmd[4 * i + c] = 1.0f / fmaxf(aligned, 1e-4f);
        }
    }
    red[threadIdx.x] = acc;
    __syncthreads();
    for (int w = 128; w > 0; w >>= 1) {
        if (threadIdx.x < w) red[threadIdx.x] += red[threadIdx.x + w];
        __syncthreads();
    }
    if (threadIdx.x == 0) blockSums[blockIdx.x] = red[0];
}

// Final: fixed-order reduce of block sums -> total loss (deterministic)
__global__ __launch_bounds__(256) void k_final(
    const float* __restrict__ blockSums, int nBlocks,
    const unsigned* __restrict__ nValid, float* __restrict__ out0)
{
    __shared__ float red[256];
    float acc = 0.0f;
    for (int i = threadIdx.x; i < nBlocks; i += 256) acc += blockSums[i];
    red[threadIdx.x] = acc;
    __syncthreads();
    for (int w = 128; w > 0; w >>= 1) {
        if (threadIdx.x < w) red[threadIdx.x] += red[threadIdx.x + w];
        __syncthreads();
    }
    if (threadIdx.x == 0) {
        unsigned n = *nValid;
        float l1 = red[0] / (float)max(n, 1u);
        float total = 0.5f * l1;            // LAMBDA_L1 * l1 + LAMBDA_SSIM * 0
        if (n < 100u) total = 0.0f;
        *out0 = total;
    }
}

// ---------------------------------------------------------------------------
// Launch
// ---------------------------------------------------------------------------
extern "C" void kernel_launch(void* const* d_in, const int* in_sizes, int n_in,
                              void* d_out, int out_size, void* d_ws, size_t ws_size,
                              hipStream_t stream)
{
    const float* depth = (const float*)d_in[0];
    const float* prior = (const float*)d_in[1];
    const int HW = in_sizes[0];                 // 1024*2048

    float* out_loss = (float*)d_out;            // [0]
    float* out_tinv = out_loss + 1;             // [1 .. HW]
    float* out_pmd  = out_tinv + HW;            // [1+HW .. 1+2HW]

    // ---- workspace layout (byte offsets) ----
    char* w = (char*)d_ws;
    unsigned* nValid   = (unsigned*)(w + 0);
    float*    med      = (float*)(w + 8);
    float*    dyn      = (float*)(w + 12);
    float*    sb       = (float*)(w + 16);
    float*    tb       = (float*)(w + 20);
    unsigned* hist1    = (unsigned*)(w + 64);                 // 16KB
    unsigned* hist2    = (unsigned*)(w + 64 + NBINS * 4);     // 16KB
    float*    scales   = (float*)(w + 64 + 2 * NBINS * 4);    // 1008*4
    float*    shifts   = scales + N_ITER_PAD;
    int*      counts   = (int*)(shifts + N_ITER_PAD);
    float*    xs       = (float*)(counts + N_ITER_PAD);       // 50000*4
    float*    ys       = xs + N_SUB;
    float*    blockSums = ys + N_SUB;                         // 1024*4

    // zero: header + histograms + scales/shifts/counts region
    size_t zbytes = 64 + 2 * (size_t)NBINS * 4 + 3 * (size_t)N_ITER_PAD * 4;
    (void)hipMemsetAsync(d_ws, 0, zbytes, stream);

    const int EW_BLOCKS = 1024;

    k_pass1<<<EW_BLOCKS, 256, 0, stream>>>(depth, out_tinv, hist1, nValid, HW);
    k_quantile<<<1, 32, 0, stream>>>(hist1, nValid, med, 0);
    k_pass2<<<EW_BLOCKS, 256, 0, stream>>>(depth, med, hist2, HW);
    k_quantile<<<1, 32, 0, stream>>>(hist2, nValid, dyn, 1);

    k_sample_pairs<<<(N_ITER_PAD + 255) / 256, 256, 0, stream>>>(
        depth, prior, scales, shifts, HW);
    k_sample_sub<<<(N_SUB + 255) / 256, 256, 0, stream>>>(
        depth, prior, xs, ys, HW);

    dim3 rg(N_ITER_PAD / 16, COL_CHUNKS);        // 63 x 64 waves
    k_ransac_count<<<rg, 32, 0, stream>>>(scales, shifts, xs, ys, dyn, counts);

    k_select<<<1, 256, 0, stream>>>(scales, shifts, counts, nValid, sb, tb);

    k_pass3<<<EW_BLOCKS, 256, 0, stream>>>(
        depth, prior, sb, tb, out_pmd, blockSums, HW);
    k_final<<<1, 256, 0, stream>>>(blockSums, EW_BLOCKS, nValid, out_loss);
}